// SpeechClassificationLayer_52166672778113
// MI455X (gfx1250) — compile-verified
//
#include <hip/hip_runtime.h>

// Speech classification: per item (32768), read 6 of 133 columns over 42 frames,
// interval rules -> frame bool/score, 5-wide sliding window via cumsum diffs,
// top-2 of valid window scores. Memory-bound gather: async global->LDS double
// buffering (ASYNCcnt) is the CDNA5 data-movement path; no matmul structure,
// so WMMA is intentionally not used.

#define FRAMES 42
#define NCLS   133
#define GROUPW 5
#define NEGV   (-1e30f)

#if __has_builtin(__builtin_amdgcn_global_load_async_to_lds_b32) && \
    __has_builtin(__builtin_amdgcn_s_wait_asynccnt)
#define USE_ASYNC_LDS 1
#else
#define USE_ASYNC_LDS 0
#endif

__device__ __forceinline__ void eval_frame(float c0, float c2, float c3,
                                           float c5, float c36, float c132,
                                           int& fj, float& fs) {
  // Rule 0: c0 in [0.6,1.0] (second condition [0,1] is subsumed)
  // Rules 1-5 share c0 in [0.5,0.7]
  bool a  = (c0 >= 0.5f) & (c0 <= 0.7f);
  bool r0 = (c0 >= 0.6f) & (c0 <= 1.0f);
  bool r1 = a & (c2   >= 0.3f) & (c2   <= 0.7f);
  bool r2 = a & (c3   >= 0.2f) & (c3   <= 0.5f);
  bool r3 = a & (c5   >= 0.2f) & (c5   <= 0.4f);
  bool r4 = a & (c132 >= 0.2f) & (c132 <= 0.5f);
  bool r5 = a & (c36  >= 0.1f) & (c36  <= 0.3f);
  fj = (int)(r0 | r1 | r2 | r3 | r4 | r5);
  // First-true rule's score (argmax over booleans = first true)
  fs = r0 ? 5.0f
     : r1 ? 1.0f
     : r2 ? 1.0f
     : r3 ? 1.5f
     : r4 ? 1.0f
     : r5 ? 1.2f
     : 0.0f;
}

extern "C" __global__ __launch_bounds__(256)
void speech_cls_kernel(const float* __restrict__ x, float* __restrict__ out, int n) {
  const int tid  = threadIdx.x;
  const int item = blockIdx.x * 256 + tid;

#if USE_ASYNC_LDS
  __shared__ float stage[2][256][6];  // [buf][thread][col], conflict-free
#endif

  if (item >= n) return;

  const float* base = x + (size_t)item * (FRAMES * NCLS);

  float cs = 0.0f;            // float prefix sum of frame scores (matches jnp.cumsum)
  int   cc = 0;               // int prefix sum of frame judgements
  float csh[GROUPW];          // ring buffer of cs_{t} (registers after full unroll)
  int   cch[GROUPW];
  float top1 = NEGV, top2 = NEGV;
  int   tc = 0;

#if USE_ASYNC_LDS
  // ROCm 7.2 signature: (int* src, int* lds_dst, imm offset, imm cpol),
  // generic pointers; clang lowers to AS1/AS3 operands itself.
#define AQ(B, RP, COL, SLOT)                                              \
  __builtin_amdgcn_global_load_async_to_lds_b32(                          \
      (int*)((RP) + (COL)),                                               \
      (int*)&stage[(B)][tid][(SLOT)],                                     \
      0, 0)
  {
    const float* rp0 = base;  // frame 0 -> buffer 0
    AQ(0, rp0, 0, 0); AQ(0, rp0, 2, 1); AQ(0, rp0, 3, 2);
    AQ(0, rp0, 5, 3); AQ(0, rp0, 36, 4); AQ(0, rp0, 132, 5);
  }
#endif

#pragma unroll
  for (int t = 0; t < FRAMES; ++t) {
    float c0, c2, c3, c5, c36, c132;
#if USE_ASYNC_LDS
    const int buf = t & 1;
    if (t + 1 < FRAMES) {
      const float* rp = base + (size_t)(t + 1) * NCLS;
      const int nb = (t + 1) & 1;
      AQ(nb, rp, 0, 0); AQ(nb, rp, 2, 1); AQ(nb, rp, 3, 2);
      AQ(nb, rp, 5, 3); AQ(nb, rp, 36, 4); AQ(nb, rp, 132, 5);
      // Async loads retire in order: <=6 outstanding means frame t's 6 landed.
      __builtin_amdgcn_s_wait_asynccnt(6);
    } else {
      __builtin_amdgcn_s_wait_asynccnt(0);
    }
    c0   = stage[buf][tid][0];
    c2   = stage[buf][tid][1];
    c3   = stage[buf][tid][2];
    c5   = stage[buf][tid][3];
    c36  = stage[buf][tid][4];
    c132 = stage[buf][tid][5];
#else
    const float* rp = base + (size_t)t * NCLS;
    if (t + 1 < FRAMES) {
      __builtin_prefetch(rp + NCLS, 0, 3);        // next row, cols 0..5 line
      __builtin_prefetch(rp + NCLS + 36, 0, 3);   // next row, col 36 line
      __builtin_prefetch(rp + NCLS + 132, 0, 3);  // next row, col 132 line
    }
    c0 = rp[0]; c2 = rp[2]; c3 = rp[3]; c5 = rp[5]; c36 = rp[36]; c132 = rp[132];
#endif

    int fj; float fs;
    eval_frame(c0, c2, c3, c5, c36, c132, fj, fs);

    csh[t % GROUPW] = cs;   // cs_{t} before adding frame t
    cch[t % GROUPW] = cc;
    cs += fs;
    cc += fj;

    if (t >= GROUPW - 1) {
      const int k = (t - (GROUPW - 1)) % GROUPW;  // slot holding cs_{t-4}
      float w  = cs - csh[k];                     // == scr_cs[i+5]-scr_cs[i]
      int   wc = cc - cch[k];
      bool  gj = (wc >= 3);
      float m  = gj ? w : NEGV;
      tc += gj ? 1 : 0;
      if (m > top1)      { top2 = top1; top1 = m; }
      else if (m > top2) { top2 = m; }
    }
  }

  bool final_j = (tc >= 2);
  out[item]     = final_j ? 1.0f : 0.0f;           // judgements
  out[n + item] = final_j ? (top1 + top2) : 0.0f;  // scores (sum of top-2)
}

extern "C" void kernel_launch(void* const* d_in, const int* in_sizes, int n_in,
                              void* d_out, int out_size, void* d_ws, size_t ws_size,
                              hipStream_t stream) {
  (void)n_in; (void)d_ws; (void)ws_size; (void)out_size;
  const float* x = (const float*)d_in[0];
  const int n = in_sizes[0] / (FRAMES * NCLS);   // 32768 items
  float* out = (float*)d_out;                    // [judgements(n) | scores(n)]
  const int threads = 256;
  const int blocks  = (n + threads - 1) / threads;
  speech_cls_kernel<<<blocks, threads, 0, stream>>>(x, out, n);
}